// GCNClassifier_64750926955043
// MI455X (gfx1250) — compile-verified
//
#include <hip/hip_runtime.h>
#include <math.h>

#define N_NODES 100000
#define N_EDGES 3200000
#define D_IN 128
#define D_HID 16
#define D_OUT 2
#define N_TILES (N_NODES / 16)   // 6250, exact

typedef __attribute__((ext_vector_type(2))) float v2f;
typedef __attribute__((ext_vector_type(8))) float v8f;

// ---------------- degree / normalization ----------------

__global__ void k_zero(float* __restrict__ p, int n) {
  int i = blockIdx.x * blockDim.x + threadIdx.x;
  if (i < n) p[i] = 0.0f;
}

__global__ void k_count_deg(const int* __restrict__ dst, float* __restrict__ deg) {
  int e = blockIdx.x * blockDim.x + threadIdx.x;
  if (e < N_EDGES) atomicAdd(&deg[dst[e]], 1.0f);
}

__global__ void k_dinv(float* __restrict__ deg) {
  int i = blockIdx.x * blockDim.x + threadIdx.x;
  if (i < N_NODES) deg[i] = rsqrtf(deg[i] + 1.0f);  // +1 for self loop
}

// ---------------- pack W1 into WMMA B-fragment layout ----------------
// Bp[ks][lane] = (W1[k0][n], W1[k0+1][n]) with n = lane%16, k0 = 4*ks + 2*(lane/16)

__global__ void k_bpack(const float* __restrict__ W1, float* __restrict__ Bp) {
  int t = blockIdx.x * blockDim.x + threadIdx.x;
  if (t < 32 * 32) {
    int ks = t >> 5, lane = t & 31;
    int n = lane & 15;
    int k0 = ks * 4 + ((lane >> 4) ? 2 : 0);
    Bp[2 * t + 0] = W1[k0 * D_HID + n];
    Bp[2 * t + 1] = W1[(k0 + 1) * D_HID + n];
  }
}

// ---------------- layer 1 GEMM via V_WMMA_F32_16X16X4_F32 ----------------
// One wave computes one 16x16 output tile: (16x128 slice of x) @ W1.
// Writes hs = h * dinv[row] and agg1 initialized with the self-loop term hs*dinv.

__global__ void k_gemm1(const float* __restrict__ x, const float* __restrict__ Bp,
                        const float* __restrict__ dinv, float* __restrict__ hs,
                        float* __restrict__ agg1) {
  int lane = threadIdx.x & 31;
  int wave = threadIdx.x >> 5;
  int tile = blockIdx.x * (blockDim.x >> 5) + wave;
  if (tile >= N_TILES) return;  // wave-uniform: EXEC all-ones past here

  int m = lane & 15;
  int koff = (lane >> 4) * 2;                      // lanes 16-31 hold K=2,3
  const float* xr = x + (size_t)(tile * 16 + m) * D_IN + koff;
  const v2f* bp = (const v2f*)Bp;

  v8f c = {0.f, 0.f, 0.f, 0.f, 0.f, 0.f, 0.f, 0.f};
#pragma unroll
  for (int ks = 0; ks < 32; ++ks) {
    v2f a = *(const v2f*)(xr + ks * 4);
    v2f b = bp[ks * 32 + lane];
    c = __builtin_amdgcn_wmma_f32_16x16x4_f32(
        /*neg_a=*/false, a, /*neg_b=*/false, b,
        /*c_mod=*/(short)0, c, /*reuse_a=*/false, /*reuse_b=*/false);
  }

  int n = lane & 15;
  int mbase = (lane >> 4) * 8;                     // C/D: VGPR v -> M = v (+8 upper half)
#pragma unroll
  for (int v = 0; v < 8; ++v) {
    int r = tile * 16 + mbase + v;
    float di = dinv[r];
    float hv = c[v] * di;                          // pre-scale by dinv[src]
    hs[(size_t)r * D_HID + n] = hv;
    agg1[(size_t)r * D_HID + n] = hv * di;         // self-loop: h * dinv^2
  }
}

// ---------------- layer 1 edge scatter: 4 threads per edge ----------------

__global__ void k_scatter1(const int* __restrict__ src, const int* __restrict__ dst,
                           const float* __restrict__ hs, const float* __restrict__ dinv,
                           float* __restrict__ agg1) {
  long long gid = (long long)blockIdx.x * blockDim.x + threadIdx.x;
  if (gid >= (long long)N_EDGES * 4) return;
  int e = (int)(gid >> 2);
  int q = (int)(gid & 3);
  int s = src[e], d = dst[e];
  float w = dinv[d];
  float4 msg = *(const float4*)(hs + (size_t)s * D_HID + q * 4);
  float* ap = agg1 + (size_t)d * D_HID + q * 4;
  atomicAdd(ap + 0, msg.x * w);
  atomicAdd(ap + 1, msg.y * w);
  atomicAdd(ap + 2, msg.z * w);
  atomicAdd(ap + 3, msg.w * w);
}

// ---------------- ReLU + bias + layer2 16->2 matvec + self loop ----------------

__global__ void k_layer2(const float* __restrict__ agg1, const float* __restrict__ b1,
                         const float* __restrict__ W2, const float* __restrict__ dinv,
                         float* __restrict__ ts, float* __restrict__ agg2) {
  int i = blockIdx.x * blockDim.x + threadIdx.x;
  if (i >= N_NODES) return;
  const float* row = agg1 + (size_t)i * D_HID;
  float t0 = 0.f, t1 = 0.f;
#pragma unroll
  for (int j = 0; j < D_HID; ++j) {
    float hj = fmaxf(row[j] + b1[j], 0.0f);
    t0 += hj * W2[j * 2 + 0];
    t1 += hj * W2[j * 2 + 1];
  }
  float di = dinv[i];
  float s0 = t0 * di, s1 = t1 * di;
  ts[2 * i + 0] = s0;
  ts[2 * i + 1] = s1;
  agg2[2 * i + 0] = s0 * di;  // self loop
  agg2[2 * i + 1] = s1 * di;
}

// ---------------- layer 2 edge scatter: 1 thread per edge ----------------

__global__ void k_scatter2(const int* __restrict__ src, const int* __restrict__ dst,
                           const float* __restrict__ ts, const float* __restrict__ dinv,
                           float* __restrict__ agg2) {
  int e = blockIdx.x * blockDim.x + threadIdx.x;
  if (e >= N_EDGES) return;
  int s = src[e], d = dst[e];
  float w = dinv[d];
  float2 m = *(const float2*)(ts + (size_t)s * 2);
  atomicAdd(&agg2[2 * d + 0], m.x * w);
  atomicAdd(&agg2[2 * d + 1], m.y * w);
}

// ---------------- bias + log_softmax ----------------

__global__ void k_final(const float* __restrict__ agg2, const float* __restrict__ b2,
                        float* __restrict__ out) {
  int i = blockIdx.x * blockDim.x + threadIdx.x;
  if (i >= N_NODES) return;
  float z0 = agg2[2 * i + 0] + b2[0];
  float z1 = agg2[2 * i + 1] + b2[1];
  float mm = fmaxf(z0, z1);
  float lse = mm + logf(expf(z0 - mm) + expf(z1 - mm));
  out[2 * i + 0] = z0 - lse;
  out[2 * i + 1] = z1 - lse;
}

extern "C" void kernel_launch(void* const* d_in, const int* in_sizes, int n_in,
                              void* d_out, int out_size, void* d_ws, size_t ws_size,
                              hipStream_t stream) {
  const float* x  = (const float*)d_in[0];   // [N, 128]
  const float* W1 = (const float*)d_in[1];   // [128, 16]
  const float* b1 = (const float*)d_in[2];   // [16]
  const float* W2 = (const float*)d_in[3];   // [16, 2]
  const float* b2 = (const float*)d_in[4];   // [2]
  const int*  ei  = (const int*)d_in[5];     // [2, E]
  const int* src = ei;
  const int* dst = ei + N_EDGES;
  float* out = (float*)d_out;

  // workspace layout (floats)
  float* w    = (float*)d_ws;
  float* deg  = w;                               // N   (becomes dinv in place)
  float* Bp   = deg + N_NODES;                   // 2048 packed W1 fragments
  float* hs   = Bp + 2048;                       // N*16  h * dinv[src]
  float* agg1 = hs + (size_t)N_NODES * D_HID;    // N*16
  float* ts   = agg1 + (size_t)N_NODES * D_HID;  // N*2   t * dinv[src]
  float* agg2 = ts + (size_t)N_NODES * D_OUT;    // N*2

  const int T = 256;
  k_zero<<<(N_NODES + T - 1) / T, T, 0, stream>>>(deg, N_NODES);
  k_count_deg<<<(N_EDGES + T - 1) / T, T, 0, stream>>>(dst, deg);
  k_dinv<<<(N_NODES + T - 1) / T, T, 0, stream>>>(deg);
  k_bpack<<<4, T, 0, stream>>>(W1, Bp);

  // 8 waves (tiles) per 256-thread block
  k_gemm1<<<(N_TILES + 7) / 8, T, 0, stream>>>(x, Bp, deg, hs, agg1);

  long long work1 = (long long)N_EDGES * 4;
  k_scatter1<<<(int)((work1 + T - 1) / T), T, 0, stream>>>(src, dst, hs, deg, agg1);

  k_layer2<<<(N_NODES + T - 1) / T, T, 0, stream>>>(agg1, b1, W2, deg, ts, agg2);
  k_scatter2<<<(N_EDGES + T - 1) / T, T, 0, stream>>>(src, dst, ts, deg, agg2);
  k_final<<<(N_NODES + T - 1) / T, T, 0, stream>>>(agg2, b2, out);
}